// Blur_40621800685871
// MI455X (gfx1250) — compile-verified
//
#include <hip/hip_runtime.h>
#include <stdint.h>

// Blur (StyleGAN2 upfirdn2d, up=down=1, pad=(2,1), kernel outer([1,3,3,1])/64)
// x: (8,128,256,256) f32 -> out: (8,128,256,256) f32.
// Memory-bound: 512 MiB round trip => ~23us floor @ 23.3 TB/s. Separable VALU
// filter (8 FMA/out ~ 1 GFLOP) hides under that; a WMMA formulation would cost
// ~5.4 TFLOP f32 and exceed the memory floor, so we use the CDNA5 async
// global->LDS data-mover path (ASYNCcnt) instead of matrix ops.

typedef __attribute__((ext_vector_type(4))) float f32x4;

constexpr int W  = 256;
constexpr int H  = 256;
constexpr int NPLANES = 8 * 128;

constexpr int TW = 64;   // tile width  (output cols per WG)
constexpr int TH = 32;   // tile height (output rows per WG)
constexpr int LROWS  = TH + 3;       // staged rows: halo 2 above, 1 below
constexpr int LCOLS  = 72;           // staged cols: tileX0-4 .. tileX0+67 (16B aligned chunks)
constexpr int CHUNKS = LCOLS / 4;    // 18 float4 chunks per staged row
constexpr int NTHREADS = 256;        // 8 waves (wave32)

#define T0 0.125f   // 1/8
#define T1 0.375f   // 3/8  (separable taps; symmetric kernel => flip == identity)

// Async global->LDS 16B copy. VGLOBAL async encoding, GV mode:
//   global_load_async_to_lds_b128 vdst(LDS byte addr), v[addr:addr+1], off
// Low 32 bits of a generic LDS pointer are exactly the LDS byte address
// (addrspacecast AS3->generic puts the aperture in addr[63:32]).
static __device__ __forceinline__ void async_copy_b128(const void* gp, void* lp) {
  asm volatile("global_load_async_to_lds_b128 %0, %1, off"
               :
               : "v"((uint32_t)(uintptr_t)lp), "v"(gp)
               : "memory");
}

static __device__ __forceinline__ void wait_asynccnt0() {
#if __has_builtin(__builtin_amdgcn_s_wait_asynccnt)
  __builtin_amdgcn_s_wait_asynccnt(0);
#else
  asm volatile("s_wait_asynccnt 0" ::: "memory");
#endif
}

__global__ __launch_bounds__(NTHREADS)
void blur_upfirdn_kernel(const float* __restrict__ x, float* __restrict__ out) {
  __shared__ alignas(16) float s_in [LROWS * LCOLS];  // raw input patch (+zeros at borders)
  __shared__ alignas(16) float s_tmp[TH    * LCOLS];  // vertically filtered rows

  const int tid    = threadIdx.x;
  const int tileX0 = blockIdx.x * TW;
  const int tileY0 = blockIdx.y * TH;
  const size_t planeOff = (size_t)blockIdx.z * (size_t)(H * W);
  const float* __restrict__ xp = x   + planeOff;
  float* __restrict__       op = out + planeOff;

  // ---- Stage: global -> LDS via async data mover ---------------------------
  // Patch rows gr = tileY0-2 .. tileY0+TH, cols gc = tileX0-4 .. tileX0+67.
  // Chunk col bases are multiples of 4 and 256 % 4 == 0, so each 16B chunk is
  // either fully inside [0,256) or fully outside -> zero-fill (zero padding).
  for (int i = tid; i < LROWS * CHUNKS; i += NTHREADS) {
    const int r  = i / CHUNKS;
    const int ck = i - r * CHUNKS;
    const int gr = tileY0 - 2 + r;
    const int gc = tileX0 - 4 + ck * 4;
    float* lp = &s_in[r * LCOLS + ck * 4];
    if ((unsigned)gr < (unsigned)H && (unsigned)gc < (unsigned)W) {
      async_copy_b128((const void*)(xp + (size_t)gr * W + gc), (void*)lp);
    } else {
      *(f32x4*)lp = (f32x4)0.0f;   // ds store; completes via DScnt at barrier
    }
  }
  wait_asynccnt0();   // all of this wave's async LDS writes committed
  __syncthreads();    // + compiler-inserted dscnt wait covers the zero-fill

  // ---- Pass 1: vertical 4-tap, f32x4, LDS -> LDS ---------------------------
  // s_tmp[r][c] = sum_a tV[a] * x[tileY0+r-2+a][tileX0-4+c]
  for (int i = tid; i < TH * CHUNKS; i += NTHREADS) {
    const int r  = i / CHUNKS;
    const int c4 = (i - r * CHUNKS) * 4;
    const f32x4 a0 = *(const f32x4*)&s_in[(r + 0) * LCOLS + c4];
    const f32x4 a1 = *(const f32x4*)&s_in[(r + 1) * LCOLS + c4];
    const f32x4 a2 = *(const f32x4*)&s_in[(r + 2) * LCOLS + c4];
    const f32x4 a3 = *(const f32x4*)&s_in[(r + 3) * LCOLS + c4];
    *(f32x4*)&s_tmp[r * LCOLS + c4] = T0 * (a0 + a3) + T1 * (a1 + a2);
  }
  __syncthreads();

  // ---- Pass 2: horizontal 4-tap, write-once NT stores to global ------------
  // out row needs s_tmp[r][co+2 .. co+8]; read 3 aligned float4s.
  for (int i = tid; i < TH * (TW / 4); i += NTHREADS) {
    const int r  = i >> 4;
    const int co = (i & 15) * 4;     // output col within tile (multiple of 4)
    const float* t = &s_tmp[r * LCOLS + co];
    const f32x4 b0 = *(const f32x4*)(t + 0);
    const f32x4 b1 = *(const f32x4*)(t + 4);
    const f32x4 b2 = *(const f32x4*)(t + 8);
    const float a2 = b0.z, a3 = b0.w;
    const float a4 = b1.x, a5 = b1.y, a6 = b1.z, a7 = b1.w;
    const float a8 = b2.x;
    f32x4 o;
    o.x = T0 * (a2 + a5) + T1 * (a3 + a4);
    o.y = T0 * (a3 + a6) + T1 * (a4 + a5);
    o.z = T0 * (a4 + a7) + T1 * (a5 + a6);
    o.w = T0 * (a5 + a8) + T1 * (a6 + a7);
    float* gp = op + (size_t)(tileY0 + r) * W + (tileX0 + co);
    __builtin_nontemporal_store(o, (f32x4*)gp);
  }
}

extern "C" void kernel_launch(void* const* d_in, const int* in_sizes, int n_in,
                              void* d_out, int out_size, void* d_ws, size_t ws_size,
                              hipStream_t stream) {
  (void)in_sizes; (void)n_in; (void)d_ws; (void)ws_size; (void)out_size;
  const float* x = (const float*)d_in[0];
  // d_in[1] is the 4x4 blur kernel: fixed normalized outer([1,3,3,1]);
  // separable taps {1,3,3,1}/8 hardcoded exactly (0.125/0.375 exact in f32).
  float* out = (float*)d_out;
  dim3 grid(W / TW, H / TH, NPLANES);   // (4, 8, 1024)
  blur_upfirdn_kernel<<<grid, dim3(NTHREADS), 0, stream>>>(x, out);
}